// EDMLoss_7267084665077
// MI455X (gfx1250) — compile-verified
//
#include <hip/hip_runtime.h>
#include <hip/hip_bf16.h>

// Problem constants (fixed by the reference's setup_inputs).
#define BB   32          // batch
#define DD   128         // feature dim
#define TT   1024        // time
#define KK   512         // codewords
#define ROWS (BB * TT)   // 32768 (b,t) pairs
#define NREC (32 * 8 * 1024)  // Xhat/X elements
#define MPAD 132         // padded LDS row stride (dwords): 128 + 4, 16B aligned

typedef __attribute__((ext_vector_type(2))) float v2f;
typedef __attribute__((ext_vector_type(8))) float v8f;
typedef __attribute__((ext_vector_type(4))) unsigned int v4u;
typedef __attribute__((ext_vector_type(4))) int v4i;
typedef __attribute__((ext_vector_type(8))) int v8i;

// Feature detection ---------------------------------------------------------
#if defined(__has_builtin)
#if __has_builtin(__builtin_amdgcn_wmma_f32_16x16x4_f32)
#define HAVE_WMMA_F32X4 1
#endif
#if __has_builtin(__builtin_amdgcn_tensor_load_to_lds) && \
    __has_builtin(__builtin_amdgcn_s_wait_tensorcnt)
#define HAVE_TDM 1
#endif
#endif

// ---------------------------------------------------------------- utilities
__device__ __forceinline__ float wave_reduce_add(float v) {
    // wave32 reduction
    #pragma unroll
    for (int off = 16; off > 0; off >>= 1) v += __shfl_down(v, off, 32);
    return v;
}

// ---------------------------------------------------------------- init
__global__ void k_init(float* accs) {
    if (threadIdx.x < 8) accs[threadIdx.x] = 0.0f;
}

// ---------------------------------------------------------------- loss_rec
__global__ void k_rec(const float* __restrict__ Xhat, const float* __restrict__ X,
                      float* __restrict__ acc) {
    float s = 0.0f;
    for (int i = blockIdx.x * blockDim.x + threadIdx.x; i < NREC;
         i += gridDim.x * blockDim.x) {
        float d = Xhat[i] - X[i];
        s += d * d;
    }
    s = wave_reduce_add(s);
    __shared__ float ws[8];
    int lane = threadIdx.x & 31, w = threadIdx.x >> 5;
    if (lane == 0) ws[w] = s;
    __syncthreads();
    if (threadIdx.x < 8) {
        float t = ws[threadIdx.x];
        #pragma unroll
        for (int off = 4; off > 0; off >>= 1) t += __shfl_down(t, off, 8);
        if (threadIdx.x == 0) atomicAdd(acc, t);
    }
}

// ---------------------------------------------------------------- M transpose
// Mt[k*128 + d] = M[d*512 + k]; row-major [K][D] so the TDM can stream it into
// LDS contiguously (TDM cannot transpose, but it CAN pad -> k*132+d layout).
__global__ void k_mt(const float* __restrict__ M, float* __restrict__ Mt) {
    int i = blockIdx.x * blockDim.x + threadIdx.x;   // 0..65535
    int d = i & (DD - 1);
    int k = i >> 7;
    Mt[i] = M[d * KK + k];   // coalesced write, strided read (256KB total)
}

// ---------------------------------------------------------------- VQ argmin
// One thread per (b,t). Entire codebook staged into the 320KB CDNA5 LDS as
// sM[k*132 + d] (270336 B). With TDM: one wave issues tensor_load_to_lds of
// Mt with pad_interval=128dw / pad_amount=4dw, reproducing the padded layout
// in hardware and tracking completion on TENSORcnt.
__global__ void __launch_bounds__(256, 1)
k_vq(const float* __restrict__ H, const float* __restrict__ M,
     const float* __restrict__ Mt, int* __restrict__ idx) {
    __shared__ float sM[KK * MPAD];   // 270336 B of the 320KB WGP LDS

#if defined(HAVE_TDM)
    if ((threadIdx.x >> 5) == 0) {   // wave 0 issues the DMA (EXEC ignored)
        unsigned long long ga = (unsigned long long)Mt;
        unsigned lds_off = (unsigned)((unsigned long long)(void*)&sM[0]);
        v4u g0;
        g0[0] = 1u;                                    // count=1, no gather
        g0[1] = lds_off;                               // lds_addr
        g0[2] = (unsigned)(ga & 0xFFFFFFFFu);          // global_addr[31:0]
        g0[3] = (unsigned)((ga >> 32) & 0x01FFFFFFu)   // global_addr[56:32]
              | (2u << 30);                            // type=2 ("image")
        v8i g1;
        g1[0] = (int)((2u << 16)      // data_size = 4B
                    | (1u << 20)      // pad_enable
                    | (6u << 22)      // pad_interval: every 128 dwords
                    | (3u << 25));    // pad_amount: 4 dwords -> stride 132
        g1[1] = (int)((DD & 0xFFFFu) << 16);           // tensor_dim0 = 128 (lo)
        g1[2] = (int)((KK & 0xFFFFu) << 16);           // dim0 hi=0, tensor_dim1 = 512 (lo)
        g1[3] = (int)((unsigned)DD << 16);             // dim1 hi=0, tile_dim0 = 128
        g1[4] = (int)(KK);                             // tile_dim1 = 512, tile_dim2 = 0
        g1[5] = (int)(DD);                             // tensor_dim0_stride = 128
        g1[6] = 0;                                     // stride0 hi, stride1 lo
        g1[7] = 0;                                     // stride1 hi
        v4i g2 = {0, 0, 0, 0};                         // 2D: groups 2/3 unused
        v4i g3 = {0, 0, 0, 0};
#if __clang_major__ >= 23
        v8i g4 = {0, 0, 0, 0, 0, 0, 0, 0};
        __builtin_amdgcn_tensor_load_to_lds(g0, g1, g2, g3, g4, 0);
#else
        __builtin_amdgcn_tensor_load_to_lds(g0, g1, g2, g3, 0);
#endif
    }
#else
    // Fallback: manual transpose staging from M[d*512+k].
    for (int i = threadIdx.x; i < DD * KK; i += blockDim.x) {
        int d = i >> 9;
        int k = i & (KK - 1);
        sM[k * MPAD + d] = M[i];
    }
#endif

    const int gid = blockIdx.x * blockDim.x + threadIdx.x;  // 0..32767
    const int b = gid >> 10;
    const int t = gid & (TT - 1);
    const float* Hb = H + (size_t)b * (DD * TT) + t;  // H[b, d, t] = Hb[d*TT]

    __builtin_prefetch(Hb, 0, 0);  // global_prefetch_b8 (CDNA5)

    // Keep the whole 128-dim query in VGPRs (fully unrolled indexing).
    float h[DD];
    #pragma unroll
    for (int d = 0; d < DD; ++d) h[d] = Hb[(size_t)d * TT];

#if defined(HAVE_TDM)
    if ((threadIdx.x >> 5) == 0) __builtin_amdgcn_s_wait_tensorcnt(0);
#endif
    __syncthreads();

    float best = 3.4e38f;
    int bestk = 0;
    for (int k = 0; k < KK; ++k) {
        const float4* m4 = (const float4*)(sM + k * MPAD);  // 16B aligned
        float a0 = 0.0f, a1 = 0.0f, a2 = 0.0f, a3 = 0.0f;
        #pragma unroll
        for (int q = 0; q < DD / 4; ++q) {
            float4 m = m4[q];
            a0 += fabsf(h[4 * q + 0] - m.x);
            a1 += fabsf(h[4 * q + 1] - m.y);
            a2 += fabsf(h[4 * q + 2] - m.z);
            a3 += fabsf(h[4 * q + 3] - m.w);
        }
        float dist = (a0 + a1) + (a2 + a3);
        if (dist < best) { best = dist; bestk = k; }   // first-min tie break
    }
    idx[gid] = bestk;
}

// ---------------------------------------------------------------- SSQ via WMMA
// Sum of squared residuals = trace(R * R^T). Each wave owns 16 rows; loops 32
// chunks of K=4 over D=128. The 16x4 f32 A-matrix register layout (lanes 0-15:
// K=0,1; lanes 16-31: K=2,3) coincides with the 4x16 B layout of its own
// transpose, so we pass A twice and accumulate the exact f32 Gram matrix with
// V_WMMA_F32_16X16X4_F32, then pull the diagonal.
__global__ void k_ssq(const float* __restrict__ H, const float* __restrict__ M,
                      const int* __restrict__ idx, float* __restrict__ acc) {
#if defined(HAVE_WMMA_F32X4)
    const int lane = threadIdx.x & 31;
    const int wave = blockIdx.x * (blockDim.x >> 5) + (threadIdx.x >> 5);
    const int row  = wave * 16 + (lane & 15);      // lanes l and l+16 share a row
    const int b = row >> 10;
    const int t = row & (TT - 1);
    const int k = idx[row];
    const float* Hb = H + (size_t)b * (DD * TT) + t;
    const float* Mk = M + k;                        // M[d*KK + k]
    const int dsel = (lane >> 4) << 1;              // 0 (lanes 0-15) or 2 (16-31)

    v8f c = {0.f, 0.f, 0.f, 0.f, 0.f, 0.f, 0.f, 0.f};
    #pragma unroll
    for (int ch = 0; ch < DD / 4; ++ch) {
        int d0 = 4 * ch + dsel;
        v2f a;
        a.x = Hb[(size_t)d0 * TT]       - Mk[(size_t)d0 * KK];
        a.y = Hb[(size_t)(d0 + 1) * TT] - Mk[(size_t)(d0 + 1) * KK];
        // D = A * A^T + C   (exact f32 accumulation)
        c = __builtin_amdgcn_wmma_f32_16x16x4_f32(
                /*neg_a=*/false, a, /*neg_b=*/false, a,
                /*c_mod=*/(short)0, c, /*reuse_a=*/false, /*reuse_b=*/false);
    }
    // Diagonal of 16x16 C/D: VGPR v holds (M=v, N=lane) for lanes 0-15 and
    // (M=v+8, N=lane-16) for lanes 16-31 -> diag lives in lane v and lane v+24.
    float part = 0.0f;
    #pragma unroll
    for (int v = 0; v < 8; ++v) {
        if (lane == v || lane == v + 24) part += c[v];
    }
    part = wave_reduce_add(part);
    if (lane == 0) atomicAdd(acc, part);
#else
    // Scalar fallback (should not be taken on gfx1250 toolchains).
    const int gid = blockIdx.x * blockDim.x + threadIdx.x;
    const int b = gid >> 10;
    const int t = gid & (TT - 1);
    const int k = idx[gid];
    const float* Hb = H + (size_t)b * (DD * TT) + t;
    const float* Mk = M + k;
    float s = 0.0f;
    #pragma unroll 8
    for (int d = 0; d < DD; ++d) {
        float r = Hb[(size_t)d * TT] - Mk[(size_t)d * KK];
        s += r * r;
    }
    s = wave_reduce_add(s);
    if ((threadIdx.x & 31) == 0) atomicAdd(acc, s);
#endif
}

// ---------------------------------------------------------------- finalize
__global__ void k_fin(const float* __restrict__ accs,
                      const float* __restrict__ Dhat,
                      const float* __restrict__ lmbda,
                      float* __restrict__ out) {
    int lane = threadIdx.x;  // launched with 32 threads (one wave)
    float v = (lane < BB) ? Dhat[lane] : 0.0f;
    v = wave_reduce_add(v);
    if (lane == 0) {
        float loss_rec = accs[0] * (1.0f / (float)NREC);
        float loss_m   = 2.0f * accs[2] * (1.0f / (float)(BB * DD * TT));
        float loss_d   = -v * (1.0f / (float)BB);
        float lmb      = lmbda[0];
        out[0] = loss_rec + 1.0f * loss_m + 0.8f * lmb * loss_d;
        out[1] = loss_rec;
        out[2] = loss_m;
        out[3] = loss_d;
        out[4] = lmb;
    }
}

// ---------------------------------------------------------------- launch
extern "C" void kernel_launch(void* const* d_in, const int* in_sizes, int n_in,
                              void* d_out, int out_size, void* d_ws, size_t ws_size,
                              hipStream_t stream) {
    const float* Xhat  = (const float*)d_in[0];
    const float* X     = (const float*)d_in[1];
    const float* H     = (const float*)d_in[2];
    const float* M     = (const float*)d_in[3];
    const float* Dhat  = (const float*)d_in[4];
    const float* lmbda = (const float*)d_in[5];
    float* out = (float*)d_out;

    float* accs = (float*)d_ws;                    // [0]=rec ssq, [2]=vq ssq
    int*   idx  = (int*)((char*)d_ws + 256);       // 32768 ints
    float* Mt   = (float*)((char*)d_ws + 256 + ROWS * sizeof(int));  // 256KB

    k_init<<<1, 32, 0, stream>>>(accs);
    k_rec <<<256, 256, 0, stream>>>(Xhat, X, accs + 0);
    k_mt  <<<DD * KK / 256, 256, 0, stream>>>(M, Mt);
    k_vq  <<<ROWS / 256, 256, 0, stream>>>(H, M, Mt, idx);
#if defined(HAVE_WMMA_F32X4)
    k_ssq <<<ROWS / 16 / 8, 256, 0, stream>>>(H, M, idx, accs + 2);  // 8 waves/blk
#else
    k_ssq <<<ROWS / 256, 256, 0, stream>>>(H, M, idx, accs + 2);
#endif
    k_fin <<<1, 32, 0, stream>>>(accs, Dhat, lmbda, out);
}